// YOLOv1Loss_3272765080392
// MI455X (gfx1250) — compile-verified
//
#include <hip/hip_runtime.h>
#include <hip/hip_bf16.h>
#include <math.h>

// ---------------------------------------------------------------------------
// YOLOv1 loss, MI455X (gfx1250, wave32).
// Memory-bound reduction: dense 0.5*conf^2 stream (NT loads, ~61MB touched)
// + sparse per-image corrections. Hierarchical reduction:
//   wave : V_WMMA_F32_16X16X4_F32 with all-ones B (one instr for 32 lanes)
//   block: ds_add_f32 into LDS (16 LDS atomics / block)
//   grid : per-block partial -> d_ws, final 1-block kernel writes out[0]
// Zero same-address global atomics (they would serialize at L2 and dominate
// the ~3us HBM stream time).
// ---------------------------------------------------------------------------

#define S_ 7
#define B_ 2
#define C_ 80
#define CELL_F (B_ * 5 + C_)   // 90 floats per cell
#define N_ 8

#define DENSE_BLOCKS 512
#define THREADS 256

typedef float v2f __attribute__((ext_vector_type(2)));
typedef float v8f __attribute__((ext_vector_type(8)));

// Per-block sum of one fp32 partial per thread.
// Wave level via f32 WMMA: A-matrix 16x4 f32 layout puts lane L<16 at
// (M=L,K=0) and lane L>=16 at (M=L-16,K=2); with B = ones,
// D[m][n] = p_m + p_{m+16} for every n. Lanes 0..15 then hold
// sum_{m=0..7} D[m][*] in c[0..7], lanes 16..31 hold sum_{m=8..15}, so
// lane0's and lane16's c-sums add to the wave total. Those two lanes
// ds_add_f32 into an LDS accumulator; thread 0 returns the block total.
// Must be called from uniform control flow (EXEC all ones at the WMMA).
__device__ __forceinline__ float block_sum(float p) {
    __shared__ float sacc;
    if (threadIdx.x == 0) sacc = 0.0f;
    __syncthreads();

    v2f a; a.x = p; a.y = 0.0f;        // K=1 / K=3 slots zero
    v2f ones; ones.x = 1.0f; ones.y = 1.0f;
    v8f c = {};
    c = __builtin_amdgcn_wmma_f32_16x16x4_f32(
            /*neg_a=*/false, a, /*neg_b=*/false, ones,
            /*c_mod=*/(short)0, c, /*reuse_a=*/false, /*reuse_b=*/false);
    float s = c[0] + c[1] + c[2] + c[3] + c[4] + c[5] + c[6] + c[7];

    int lane = threadIdx.x & 31;
    if (lane == 0 || lane == 16) {
        atomicAdd(&sacc, s);           // ds_add_f32
    }
    __syncthreads();
    return sacc;
}

__device__ __forceinline__ float iou_f(float px1, float py1, float px2, float py2,
                                       float gx1, float gy1, float gx2, float gy2) {
    float ix1 = fmaxf(px1, gx1), iy1 = fmaxf(py1, gy1);
    float ix2 = fminf(px2, gx2), iy2 = fminf(py2, gy2);
    float inter = fmaxf(ix2 - ix1, 0.f) * fmaxf(iy2 - iy1, 0.f);
    float a1 = fmaxf(px2 - px1, 0.f) * fmaxf(py2 - py1, 0.f);
    float a2 = fmaxf(gx2 - gx1, 0.f) * fmaxf(gy2 - gy1, 0.f);
    return inter / (a1 + a2 - inter + 1e-6f);
}

// Dense part: 0.5 * sum(conf^2) over all (cell, box). Streams only the two
// confidence floats of each 360B cell with non-temporal hints (single-use
// data; keep it out of L2).
__global__ void yolo_conf_dense(const float* __restrict__ preds,
                                float* __restrict__ part, int ncells) {
    int tid = blockIdx.x * blockDim.x + threadIdx.x;
    int nthreads = gridDim.x * blockDim.x;
    float acc = 0.0f;
    for (int cell = tid; cell < ncells; cell += nthreads) {
        size_t base = (size_t)cell * CELL_F;
        float c0 = __builtin_nontemporal_load(preds + base + 4);
        float c1 = __builtin_nontemporal_load(preds + base + 9);
        acc += 0.5f * (c0 * c0 + c1 * c1);
    }
    float t = block_sum(acc);
    if (threadIdx.x == 0) part[blockIdx.x] = t;
}

// Sparse part: one thread per image. Replays the lax.scan overwrite semantics:
// last-writer-wins per (cell, best-box) slot; cls targets are a set-union of
// labels per cell; class loss only for cells containing >=1 GT center.
// Contributes:  [(conf-iou)^2 - 0.5 conf^2] + 5*coordSq  at final slots,
//               sum_c (pred_cls - cls_t)^2                at distinct cells.
__global__ void yolo_sparse(const float* __restrict__ preds,
                            const float* __restrict__ gtb,
                            const int* __restrict__ gtl,
                            float* __restrict__ part, int bsz) {
    int b = blockIdx.x * blockDim.x + threadIdx.x;
    float corr = 0.0f;
    if (b < bsz) {
        const float* Pb = preds + (size_t)b * (S_ * S_ * CELL_F);
        const float* G  = gtb + (size_t)b * (N_ * 4);
        const int*   L  = gtl + (size_t)b * N_;

        int   cid[N_], slot[N_], lab[N_];
        float iouB[N_], confB[N_], coordSq[N_];

        #pragma unroll
        for (int k = 0; k < N_; k++) {
            float gx1 = G[k * 4 + 0], gy1 = G[k * 4 + 1];
            float gx2 = G[k * 4 + 2], gy2 = G[k * 4 + 3];
            float gcx = 0.5f * (gx1 + gx2), gcy = 0.5f * (gy1 + gy2);
            float gw = gx2 - gx1, gh = gy2 - gy1;
            int gi = (int)(gcx * (float)S_); gi = gi < 0 ? 0 : (gi > S_ - 1 ? S_ - 1 : gi);
            int gj = (int)(gcy * (float)S_); gj = gj < 0 ? 0 : (gj > S_ - 1 ? S_ - 1 : gj);
            int cc = gj * S_ + gi;
            const float* cp = Pb + cc * CELL_F;

            float tx = gcx * (float)S_ - (float)gi;
            float ty = gcy * (float)S_ - (float)gj;
            float tw = sqrtf(fmaxf(gw, 0.f)), th = sqrtf(fmaxf(gh, 0.f));

            float iou[2], cf[2], csq[2];
            #pragma unroll
            for (int bk = 0; bk < B_; bk++) {
                float p0 = cp[bk * 5 + 0], p1 = cp[bk * 5 + 1];
                float p2 = cp[bk * 5 + 2], p3 = cp[bk * 5 + 3];
                float p4 = cp[bk * 5 + 4];
                float pcx = (p0 + (float)gi) * (1.0f / S_);
                float pcy = (p1 + (float)gj) * (1.0f / S_);
                float pw = p2 * p2, ph = p3 * p3;
                iou[bk] = iou_f(pcx - 0.5f * pw, pcy - 0.5f * ph,
                                pcx + 0.5f * pw, pcy + 0.5f * ph,
                                gx1, gy1, gx2, gy2);
                cf[bk] = p4;
                float d0 = p0 - tx, d1 = p1 - ty, d2 = p2 - tw, d3 = p3 - th;
                csq[bk] = d0 * d0 + d1 * d1 + d2 * d2 + d3 * d3;
            }
            int bst = (iou[1] > iou[0]) ? 1 : 0;   // argmax, ties -> 0
            cid[k] = cc;
            slot[k] = cc * B_ + bst;
            lab[k] = L[k];
            iouB[k]    = bst ? iou[1] : iou[0];
            confB[k]   = bst ? cf[1]  : cf[0];
            coordSq[k] = bst ? csq[1] : csq[0];
        }

        // conf + coord corrections at last-writer slots
        #pragma unroll
        for (int k = 0; k < N_; k++) {
            bool last = true;
            #pragma unroll
            for (int k2 = k + 1; k2 < N_; k2++)
                if (slot[k2] == slot[k]) last = false;
            if (last) {
                float cf = confB[k], io = iouB[k];
                corr += (cf - io) * (cf - io) - 0.5f * cf * cf
                        + 5.0f * coordSq[k];
            }
        }

        // class loss for distinct occupied cells
        #pragma unroll
        for (int k = 0; k < N_; k++) {
            bool first = true;
            #pragma unroll
            for (int k2 = 0; k2 < k; k2++)
                if (cid[k2] == cid[k]) first = false;
            if (first) {
                const float* cp = Pb + cid[k] * CELL_F + B_ * 5;
                float s = 0.0f;
                #pragma unroll
                for (int c = 0; c < C_; c++) { float q = cp[c]; s += q * q; }
                corr += s;  // sum pred_cls^2
                // (p - 1)^2 - p^2 = 1 - 2p for each distinct (cell, label)
                #pragma unroll
                for (int k2 = k; k2 < N_; k2++) {
                    if (cid[k2] == cid[k]) {
                        bool dup = false;
                        #pragma unroll
                        for (int k3 = k; k3 < k2; k3++)
                            if (cid[k3] == cid[k] && lab[k3] == lab[k2]) dup = true;
                        if (!dup) corr += 1.0f - 2.0f * cp[lab[k2]];
                    }
                }
            }
        }
    }
    float t = block_sum(corr);         // uniform: every thread reaches this
    if (threadIdx.x == 0) part[blockIdx.x] = t;
}

// Final: one block sums all per-block partials and writes the scalar loss.
__global__ void yolo_final(const float* __restrict__ part, int n,
                           float* __restrict__ out) {
    float acc = 0.0f;
    for (int i = threadIdx.x; i < n; i += blockDim.x) acc += part[i];
    float t = block_sum(acc);
    if (threadIdx.x == 0) out[0] = t;
}

extern "C" void kernel_launch(void* const* d_in, const int* in_sizes, int n_in,
                              void* d_out, int out_size, void* d_ws, size_t ws_size,
                              hipStream_t stream) {
    const float* preds = (const float*)d_in[0];
    const float* gtb   = (const float*)d_in[1];
    const int*   gtl   = (const int*)d_in[2];
    float* out  = (float*)d_out;
    float* part = (float*)d_ws;

    int bsz = in_sizes[0] / (S_ * S_ * CELL_F);   // 8192
    int ncells = bsz * S_ * S_;
    int sparse_blocks = (bsz + THREADS - 1) / THREADS;
    int npart = DENSE_BLOCKS + sparse_blocks;

    yolo_conf_dense<<<DENSE_BLOCKS, THREADS, 0, stream>>>(preds, part, ncells);
    yolo_sparse<<<sparse_blocks, THREADS, 0, stream>>>(preds, gtb, gtl,
                                                       part + DENSE_BLOCKS, bsz);
    yolo_final<<<1, THREADS, 0, stream>>>(part, npart, out);
}